// AttentionE_14800457302540
// MI455X (gfx1250) — compile-verified
//
#include <hip/hip_runtime.h>

// AttentionE: out[i,j] = e[i,j] * softmax_j(tanh(sum(d[i,:]) * W * e[i,j] + b[j]))
// Memory-bound (64 B/sample, ~0.8 FLOP/B): single streaming pass using the
// CDNA5 async global<->LDS b128 path (ASYNCcnt) with non-temporal hints.

#define TPB  256
#define SAMP 256

// Low 32 bits of a generic pointer into __shared__ == LDS byte offset (aperture rule).
__device__ __forceinline__ unsigned lds_off(const void* p) {
  return (unsigned)(unsigned long long)p;
}

__device__ __forceinline__ void async_ld_b128(unsigned lds, const void* g) {
  asm volatile("global_load_async_to_lds_b128 %0, %1, off th:TH_LOAD_NT"
               :: "v"(lds), "v"((unsigned long long)g) : "memory");
}

__device__ __forceinline__ void async_st_b128(const void* g, unsigned lds) {
  asm volatile("global_store_async_from_lds_b128 %0, %1, off th:TH_STORE_NT"
               :: "v"((unsigned long long)g), "v"(lds) : "memory");
}

__device__ __forceinline__ void wait_async0() {
  asm volatile("s_wait_asynccnt 0" ::: "memory");
}

__device__ __forceinline__ float fast_tanh(float x) {
#if defined(__has_builtin) && __has_builtin(__builtin_amdgcn_tanhf)
  return __builtin_amdgcn_tanhf(x);   // native v_tanh_f32 (CDNA5 TRANS op)
#else
  return tanhf(x);
#endif
}

__global__ __launch_bounds__(TPB) void attn_e_kernel(
    const float* __restrict__ dmat,   // (N,6)
    const float* __restrict__ emat,   // (N,5)
    const float* __restrict__ Wp,     // (1,)
    const float* __restrict__ bp,     // (5,)
    float* __restrict__ out,          // (N,5)
    int n)
{
  // d tile: 1536 f (6144B) | e tile: 1280 f (5120B) | o tile: 1280 f (5120B) = 16KB
  __shared__ __align__(16) float sm[SAMP * 16];
  float* smd = sm;               // byte offset 0
  float* sme = sm + SAMP * 6;    // byte offset 6144  (16B aligned)
  float* smo = sm + SAMP * 11;   // byte offset 11264 (16B aligned)

  const int tid = threadIdx.x;
  const long long base = (long long)blockIdx.x * SAMP;

  const float Wv = Wp[0];
  float bb[5];
#pragma unroll
  for (int j = 0; j < 5; ++j) bb[j] = bp[j];

  if (base + SAMP <= (long long)n) {
    // ---- async global -> LDS: straight-line, fully coalesced b128 streams ----
    const char* gd = (const char*)(dmat + base * 6);   // 6144B = 384 xfers
    const char* ge = (const char*)(emat + base * 5);   // 5120B = 320 xfers
    async_ld_b128(lds_off(&smd[tid * 4]), gd + (size_t)tid * 16);
    if (tid < 128)
      async_ld_b128(lds_off(&smd[(tid + 256) * 4]), gd + (size_t)(tid + 256) * 16);
    async_ld_b128(lds_off(&sme[tid * 4]), ge + (size_t)tid * 16);
    if (tid < 64)
      async_ld_b128(lds_off(&sme[(tid + 256) * 4]), ge + (size_t)(tid + 256) * 16);
    wait_async0();
    __syncthreads();

    // ---- per-sample compute (strides 5 & 6 are LDS-bank-conflict-free) ----
    const int di = tid * 6, ei = tid * 5;
    const float s = ((smd[di] + smd[di + 1]) + (smd[di + 2] + smd[di + 3])) +
                    (smd[di + 4] + smd[di + 5]);
    const float sw = s * Wv;
    float ev[5], av[5];
    float asum = 0.0f;
#pragma unroll
    for (int j = 0; j < 5; ++j) {
      ev[j] = sme[ei + j];
      const float t = fast_tanh(fmaf(sw, ev[j], bb[j]));  // v_tanh_f32
      av[j] = __expf(t);                                  // v_exp_f32
      asum += av[j];
    }
    const float r = 1.0f / (asum + 1e-7f);
#pragma unroll
    for (int j = 0; j < 5; ++j)
      smo[ei + j] = ev[j] * (av[j] * r);

    __syncthreads();

    // ---- async LDS -> global: straight-line coalesced b128 output stream ----
    char* go = (char*)(out + base * 5);                // 5120B = 320 xfers
    async_st_b128(go + (size_t)tid * 16, lds_off(&smo[tid * 4]));
    if (tid < 64)
      async_st_b128(go + (size_t)(tid + 256) * 16, lds_off(&smo[(tid + 256) * 4]));
    wait_async0();
  } else {
    // Scalar tail (not taken for N = 4194304, kept for robustness).
    const long long idx = base + tid;
    if (idx < (long long)n) {
      const float* dr = dmat + idx * 6;
      const float* er = emat + idx * 5;
      const float s = dr[0] + dr[1] + dr[2] + dr[3] + dr[4] + dr[5];
      const float sw = s * Wv;
      float ev[5], av[5];
      float asum = 0.0f;
#pragma unroll
      for (int j = 0; j < 5; ++j) {
        ev[j] = er[j];
        av[j] = __expf(fast_tanh(fmaf(sw, ev[j], bb[j])));
        asum += av[j];
      }
      const float r = 1.0f / (asum + 1e-7f);
      float* orow = out + idx * 5;
#pragma unroll
      for (int j = 0; j < 5; ++j) orow[j] = ev[j] * (av[j] * r);
    }
  }
}

extern "C" void kernel_launch(void* const* d_in, const int* in_sizes, int n_in,
                              void* d_out, int out_size, void* d_ws, size_t ws_size,
                              hipStream_t stream) {
  (void)n_in; (void)d_ws; (void)ws_size; (void)out_size;
  const float* dmat = (const float*)d_in[0];  // (N,6)
  const float* emat = (const float*)d_in[1];  // (N,5)
  const float* W    = (const float*)d_in[2];  // (1,)
  const float* b    = (const float*)d_in[3];  // (5,)
  float* out = (float*)d_out;                 // (N,5)

  const int n = in_sizes[0] / 6;              // samples
  const int blocks = (n + SAMP - 1) / SAMP;   // 16384 for N=4194304
  attn_e_kernel<<<blocks, TPB, 0, stream>>>(dmat, emat, W, b, out, n);
}